// EmbeddingGenePooler_1443109012239
// MI455X (gfx1250) — compile-verified
//
#include <hip/hip_runtime.h>
#include <stdint.h>

// ---------------------------------------------------------------------------
// Sorted-COO segment sum: out[seg] += emb[frag]   (memory-bound, ~1.5 GB)
// Tile reduction via V_WMMA_F32_16X16X4_F32 with a one-hot f32 A-matrix.
// Embedding tiles (16 rows x 256B, contiguous) staged to LDS with coalesced
// async b128 transfers (ASYNCcnt double buffer) when the builtin exists.
// ---------------------------------------------------------------------------

typedef float v2f __attribute__((ext_vector_type(2)));
typedef float v8f __attribute__((ext_vector_type(8)));
typedef int vi4 __attribute__((vector_size(16)));  // matches builtin param type

#if defined(__has_builtin)
#if __has_builtin(__builtin_amdgcn_wmma_f32_16x16x4_f32)
#define HAVE_WMMA_F32X4 1
#endif
#if __has_builtin(__builtin_amdgcn_global_load_async_to_lds_b128)
#define HAVE_ASYNC_LDS 1
#endif
#endif

#if defined(HAVE_ASYNC_LDS)
#if defined(__has_builtin) && __has_builtin(__builtin_amdgcn_s_wait_asynccnt)
#define WAIT_ASYNC(n) __builtin_amdgcn_s_wait_asynccnt(n)
#else
#define WAIT_ASYNC(n) asm volatile("s_wait_asynccnt " #n ::: "memory")
#endif
typedef __attribute__((address_space(1))) vi4* as1_vi4;
typedef __attribute__((address_space(3))) vi4* as3_vi4;
#endif

#define WAVES_PER_BLOCK 8
#define BLOCK_THREADS   (32 * WAVES_PER_BLOCK)
#define CHUNK           256  // fragments per wave (contiguous, sorted)
#define TILE            16   // fragments per WMMA tile (<=16 distinct segs)
#define D_EMB           64
#define ROWP            80   // padded LDS row stride (floats) -> no bank conflicts
#define TILEF           (TILE * ROWP)

#if defined(HAVE_ASYNC_LDS)
#define NBUF 2
#else
#define NBUF 1
#endif

__global__ __launch_bounds__(256) void zero_out_kernel(float4* __restrict__ out,
                                                       long n4) {
  long i = (long)blockIdx.x * blockDim.x + threadIdx.x;
  long stride = (long)gridDim.x * blockDim.x;
  float4 z; z.x = 0.f; z.y = 0.f; z.z = 0.f; z.w = 0.f;
  for (; i < n4; i += stride) out[i] = z;
}

// Flush accumulator rows 0..maxrow. Row 0 / row maxrow may be shared or
// partially flushed -> atomic add; interior rows are exclusive -> plain store.
__device__ __forceinline__ void flush_rows(v8f acc[4],
                                           const unsigned int* __restrict__ segp,
                                           float* __restrict__ out,
                                           int maxrow, int col, int half) {
#pragma unroll
  for (int m = 0; m < 8; ++m) {
    int row = m + (half << 3);
    if (row <= maxrow) {
      unsigned int seg = segp[row];
      float* dst = out + (size_t)seg * D_EMB + col;
      bool at = (row == 0) || (row == maxrow);
#pragma unroll
      for (int s = 0; s < 4; ++s) {
        float v = acc[s][m];
        if (at) atomicAdd(dst + s * 16, v);
        else dst[s * 16] = v;
      }
    }
  }
}

struct TileRegs { float4 v[8]; };

// Linear element q = i*32+lane covers the tile's contiguous 4KB: row = q>>4,
// quad = q&15; clamp the row for the ragged tail (A column is 0 there).
__device__ __forceinline__ const float* tile_gaddr(const float* __restrict__ emb,
                                                   long f0, long end,
                                                   int i, int lane) {
  int q = i * 32 + lane;
  long fr = f0 + (q >> 4);
  if (fr >= end) fr = end - 1;
  return emb + fr * (long)D_EMB + (q & 15) * 4;
}

#if defined(HAVE_ASYNC_LDS)
__device__ __forceinline__ void stage_async(const float* __restrict__ emb,
                                            long f0, long end,
                                            float* lrow, int lane) {
#pragma unroll
  for (int i = 0; i < 8; ++i) {
    int q = i * 32 + lane;
    const float* g = tile_gaddr(emb, f0, end, i, lane);
    float* l = lrow + (q >> 4) * ROWP + (q & 15) * 4;
    __builtin_amdgcn_global_load_async_to_lds_b128(
        reinterpret_cast<as1_vi4>(reinterpret_cast<uintptr_t>(g)),
        reinterpret_cast<as3_vi4>(reinterpret_cast<uintptr_t>(l)), 0, 0);
  }
}
#else
__device__ __forceinline__ void load_tile(const float* __restrict__ emb,
                                          long f0, long end, int lane,
                                          TileRegs& t) {
#pragma unroll
  for (int i = 0; i < 8; ++i)
    t.v[i] = *(const float4*)tile_gaddr(emb, f0, end, i, lane);
}
#endif

__global__ __launch_bounds__(BLOCK_THREADS) void seg_pool_wmma_kernel(
    const float* __restrict__ emb,
    const unsigned int* __restrict__ fidx2,  // low dword of each int64 index
    float* __restrict__ out,
    long n_frag) {
  __shared__ float lbuf[WAVES_PER_BLOCK][NBUF][TILEF];
  __shared__ unsigned int seg_lds[WAVES_PER_BLOCK][16];

  const int lane = threadIdx.x & 31;
  const int wave = threadIdx.x >> 5;
  const int col  = lane & 15;  // C/D column; also A-row (M) for this lane
  const int half = lane >> 4;

  const long chunkId = (long)blockIdx.x * WAVES_PER_BLOCK + wave;
  const long base = chunkId * CHUNK;
  if (base >= n_frag) return;
  const long end = (base + CHUNK < n_frag) ? (base + CHUNK) : n_frag;

  unsigned int* segp = seg_lds[wave];

  v8f acc[4];
  v8f zero8 = {0.f, 0.f, 0.f, 0.f, 0.f, 0.f, 0.f, 0.f};
#pragma unroll
  for (int s = 0; s < 4; ++s) acc[s] = zero8;

  int slot_carry = -1;
  unsigned int prev_last = 0xFFFFFFFFu;

#if defined(HAVE_ASYNC_LDS)
  int buf = 0;
  stage_async(emb, base, end, &lbuf[wave][0][0], lane);
#else
  TileRegs tr;
  load_tile(emb, base, end, lane, tr);
#endif

  for (long f0 = base; f0 < end; f0 += TILE) {
    const bool more = (f0 + TILE) < end;

#if defined(HAVE_ASYNC_LDS)
    if (more) stage_async(emb, f0 + TILE, end, &lbuf[wave][buf ^ 1][0], lane);
    const int cb = buf;
#else
    // park current tile in LDS, then immediately start next tile's global loads
#pragma unroll
    for (int i = 0; i < 8; ++i) {
      int q = i * 32 + lane;
      *(float4*)&lbuf[wave][0][(q >> 4) * ROWP + (q & 15) * 4] = tr.v[i];
    }
    TileRegs trn = tr;
    if (more) load_tile(emb, f0 + TILE, end, lane, trn);
    const int cb = 0;
#endif

    // ---- indices, run boundaries, prefix-sum -> per-fragment slot ----
    long f = f0 + col;
    bool valid = f < end;
    long fc = valid ? f : (end - 1);
    unsigned int idx32 = fidx2[(size_t)fc * 2];

    {  // GL2 stream-ahead hint
      long pf = f0 + 2 * TILE;
      if (pf < end) __builtin_prefetch(emb + pf * D_EMB + (size_t)lane * 16, 0, 1);
    }

    unsigned int idx_prev = __shfl(idx32, lane - 1);
    unsigned int pv = (col == 0) ? prev_last : idx_prev;
    int srel = (valid && idx32 != pv) ? 1 : 0;
#pragma unroll
    for (int off = 1; off < 16; off <<= 1) {
      int t = __shfl(srel, lane - off);
      if (col >= off) srel += t;
    }
    int slot = slot_carry + srel;

    int slot15 = __shfl(slot, 15);
    if (slot15 > 15) {  // >16 live runs: flush and rebase
      flush_rows(acc, segp, out, slot_carry, col, half);
#pragma unroll
      for (int s = 0; s < 4; ++s) acc[s] = zero8;
      int slot0 = __shfl(slot, 0);
      slot -= slot0;  // carried/partial run -> row 0 (always atomic later)
    }

    int mslot = valid ? slot : -1;
    if (valid && lane < 16) segp[slot] = idx32;

#if defined(HAVE_ASYNC_LDS)
    if (more) { WAIT_ASYNC(8); } else { WAIT_ASYNC(0); }
    asm volatile("" ::: "memory");
#endif

    // ---- WMMA: one-hot A (16x4) x LDS-staged B (4x16) per 16-dim slice ----
    const float* lt = &lbuf[wave][cb][0];
#pragma unroll
    for (int g = 0; g < 4; ++g) {
      int j0 = g * 4 + 2 * half;  // A vgpr0: K={0},{2}; vgpr1: K={1},{3}
      int sj0 = __shfl(mslot, j0);
      int sj1 = __shfl(mslot, j0 + 1);
      v2f a;
      a.x = (sj0 == col) ? 1.0f : 0.0f;
      a.y = (sj1 == col) ? 1.0f : 0.0f;
      int r0 = j0 * ROWP + col;
      int r1 = r0 + ROWP;
#pragma unroll
      for (int s = 0; s < 4; ++s) {
        v2f b;
        b.x = lt[r0 + s * 16];
        b.y = lt[r1 + s * 16];
#if defined(HAVE_WMMA_F32X4)
        acc[s] = __builtin_amdgcn_wmma_f32_16x16x4_f32(
            false, a, false, b, (short)0, acc[s], false, false);
#else
#pragma unroll
        for (int m = 0; m < 8; ++m) {
          int row = m + (half << 3);
          acc[s][m] += ((sj0 == row) ? b.x : 0.f) + ((sj1 == row) ? b.y : 0.f);
        }
#endif
      }
    }

    slot_carry = __shfl(slot, 15);
    prev_last = __shfl(idx32, 15);
#if defined(HAVE_ASYNC_LDS)
    buf ^= 1;
#else
    if (more) tr = trn;
#endif
  }

  if (slot_carry >= 0) flush_rows(acc, segp, out, slot_carry, col, half);
}

extern "C" void kernel_launch(void* const* d_in, const int* in_sizes, int n_in,
                              void* d_out, int out_size, void* d_ws, size_t ws_size,
                              hipStream_t stream) {
  const float* emb = (const float*)d_in[0];
  const unsigned int* fidx2 = (const unsigned int*)d_in[1];  // int64 low dwords
  float* out = (float*)d_out;

  long n_frag = (long)in_sizes[0] / D_EMB;
  long n4 = (long)out_size / 4;

  zero_out_kernel<<<2048, 256, 0, stream>>>((float4*)out, n4);

  long nchunks = (n_frag + CHUNK - 1) / CHUNK;
  long nblocks = (nchunks + WAVES_PER_BLOCK - 1) / WAVES_PER_BLOCK;
  seg_pool_wmma_kernel<<<dim3((unsigned)nblocks), dim3(BLOCK_THREADS), 0, stream>>>(
      emb, fidx2, out, n_frag);
}